// SparseMaxPool_8194797600857
// MI455X (gfx1250) — compile-verified
//
#include <hip/hip_runtime.h>
#include <math.h>

typedef __attribute__((ext_vector_type(2))) float v2f;
typedef __attribute__((ext_vector_type(4))) float v4f;
typedef __attribute__((ext_vector_type(8))) float v8f;

#define NSEQ 64
#define DDIM 512
#define BATCH 16
#define OCAT 1024   // 512 (m_k via w_v) + 512 (m_q via w_c[:512])

// ---------------------------------------------------------------------------
// Kernel A: m_kq[b, n, o] = sum_d x[b,d,n] * W[o,d] + bias[o]
//   o <  512 : W = w_v, bias = b_v  (this is m_k)
//   o >= 512 : W = w_c (rows 0..511), bias = b_c[:512]  (this is m_q)
// One wave (32 lanes) computes one 16x16 output tile with v_wmma_f32_16x16x4_f32.
// ---------------------------------------------------------------------------
__global__ __launch_bounds__(256) void gemm_kq(
    const float* __restrict__ x, const float* __restrict__ w_c,
    const float* __restrict__ b_c, const float* __restrict__ w_v,
    const float* __restrict__ b_v, float* __restrict__ mkq)
{
  int wave = (int)((blockIdx.x * 256u + threadIdx.x) >> 5);  // 0..4095
  int lane = threadIdx.x & 31;
  int h    = lane >> 4;        // K half (0: K=0,1  1: K=2,3)
  int l16  = lane & 15;
  int ot = wave & 63;          // 64 o-tiles of 16
  int mt = (wave >> 6) & 3;    // 4 n-tiles of 16
  int b  = wave >> 8;          // 16 batches
  int obase = ot * 16, nbase = mt * 16;

  const float* Wp;
  const float* bp;
  if (obase < 512) { Wp = w_v + (size_t)obase * DDIM;         bp = b_v + obase; }
  else             { Wp = w_c + (size_t)(obase - 512) * DDIM; bp = b_c + (obase - 512); }

  // A[m][k] = x[b, k, nbase+m]   (m_feats = x transposed)
  const float* xa = x + (size_t)b * DDIM * NSEQ + (nbase + l16);
  // B[k][n] = W[obase+n][k]  -> contiguous float2 along k
  const float* wb = Wp + (size_t)l16 * DDIM;

  v8f c = {0.f,0.f,0.f,0.f,0.f,0.f,0.f,0.f};
  for (int k0 = 0; k0 < DDIM; k0 += 4) {
    int kk = k0 + 2 * h;
    v2f a, bf;
    a.x = xa[(size_t)kk * NSEQ];
    a.y = xa[(size_t)(kk + 1) * NSEQ];
    bf  = *(const v2f*)(wb + kk);
    c = __builtin_amdgcn_wmma_f32_16x16x4_f32(false, a, false, bf,
                                              (short)0, c, false, false);
  }
  float bias = bp[l16];
  float* outp = mkq + (size_t)b * NSEQ * OCAT + obase + l16;
  for (int v = 0; v < 8; ++v) {
    int nrow = nbase + v + 8 * h;                 // C/D layout: M = v + 8*half
    outp[(size_t)nrow * OCAT] = c[v] + bias;
  }
}

// ---------------------------------------------------------------------------
// Kernel B: logits[b,n,m] = 0.125 * sum_d m_k[b,n,d]*m_q[b,m,d]; softmax over m.
// One block per batch, 4 waves; wave w owns n-rows [16w,16w+16).
// ---------------------------------------------------------------------------
__global__ __launch_bounds__(128) void attn_softmax(
    const float* __restrict__ mkq, float* __restrict__ wout)
{
  __shared__ float logits[NSEQ][NSEQ + 1];
  int b    = blockIdx.x;
  int lane = threadIdx.x & 31;
  int wave = threadIdx.x >> 5;     // 0..3
  int h    = lane >> 4;
  int l16  = lane & 15;
  int nbase = wave * 16;

  const float* base = mkq + (size_t)b * NSEQ * OCAT;
  const float* ka = base + (size_t)(nbase + l16) * OCAT;        // m_k row (A)
  for (int mt = 0; mt < 4; ++mt) {
    const float* qb = base + (size_t)(mt * 16 + l16) * OCAT + 512; // m_q row (B col)
    v8f c = {0.f,0.f,0.f,0.f,0.f,0.f,0.f,0.f};
    for (int k0 = 0; k0 < DDIM; k0 += 4) {
      int kk = k0 + 2 * h;
      v2f a  = *(const v2f*)(ka + kk);
      v2f bf = *(const v2f*)(qb + kk);
      c = __builtin_amdgcn_wmma_f32_16x16x4_f32(false, a, false, bf,
                                                (short)0, c, false, false);
    }
    for (int v = 0; v < 8; ++v)
      logits[nbase + v + 8 * h][mt * 16 + l16] = c[v] * 0.125f;
  }
  __syncthreads();

  if (threadIdx.x < NSEQ) {
    int r = threadIdx.x;
    float mx = logits[r][0];
    for (int m = 1; m < NSEQ; ++m) mx = fmaxf(mx, logits[r][m]);
    float s = 0.f;
    for (int m = 0; m < NSEQ; ++m) {
      float e = __expf(logits[r][m] - mx);
      logits[r][m] = e;
      s += e;
    }
    float inv = 1.f / s;
    float* wp = wout + (size_t)b * NSEQ * NSEQ + (size_t)r * NSEQ;
    for (int m = 0; m < NSEQ; ++m) wp[m] = logits[r][m] * inv;
  }
}

// ---------------------------------------------------------------------------
// Kernel C: streaming writer of boundary/local/content (the ~400 MB).
// Block = (batch b, chunk of 8 d-values). LDS caches (1+w[b]) once per block,
// then per d: the x row and a running-max table rm[i][j] = max(x[i..j]).
//   boundary[i,j] = (x_i+x_j)/2             (masked), x_i on diag
//   local[i,j]    = (x_i+x_j+0.5*x_mid)/2.5 (masked), x_i on diag
//   content[i,j]  = max(x[i..j])            (masked incl. diag)
// all scaled by (1 + softmax weight), zeros elsewhere.
// Each thread produces 4 consecutive j (same row i) -> global_store_b128,
// non-temporal (output stream >> L2, zero reuse).
// ---------------------------------------------------------------------------
__global__ __launch_bounds__(256) void build_maps(
    const float* __restrict__ x, const float* __restrict__ w,
    float* __restrict__ out)
{
  __shared__ __align__(16) float wp[NSEQ * NSEQ];
  __shared__ __align__(16) float rm[NSEQ * NSEQ];
  __shared__ float xr[NSEQ];

  int b      = blockIdx.x >> 6;
  int dchunk = blockIdx.x & 63;
  int t      = threadIdx.x;

  const float* wsrc = w + (size_t)b * NSEQ * NSEQ;
  for (int r = 0; r < 16; ++r) wp[t + 256 * r] = 1.0f + wsrc[t + 256 * r];

  const size_t mapstride = (size_t)BATCH * DDIM * NSEQ * NSEQ; // 33,554,432
  float* outB = out;
  float* outL = out + mapstride;
  float* outC = out + 2 * mapstride;

  for (int dd = 0; dd < 8; ++dd) {
    int d = dchunk * 8 + dd;
    __syncthreads();
    if (t < NSEQ) xr[t] = x[((size_t)b * DDIM + d) * NSEQ + t];
    __syncthreads();
    if (t < NSEQ) {                 // running max per row i = t
      float m = xr[t];
      rm[t * NSEQ + t] = m;
      for (int j = t + 1; j < NSEQ; ++j) {
        m = fmaxf(m, xr[j]);
        rm[t * NSEQ + j] = m;
      }
    }
    __syncthreads();

    size_t obase = ((size_t)b * DDIM + d) * (NSEQ * NSEQ);
    for (int r = 0; r < 4; ++r) {
      int g  = t + 256 * r;          // group of 4 elements, 0..1023
      int i  = g >> 4;               // row
      int j0 = (g & 15) * 4;         // first column of the 4
      int idx = i * NSEQ + j0;

      v4f wv = *(const v4f*)(wp + idx);
      v4f vb = {0.f,0.f,0.f,0.f};
      v4f vl = {0.f,0.f,0.f,0.f};
      v4f vc = {0.f,0.f,0.f,0.f};

      float xi = xr[i];
      #pragma unroll
      for (int e = 0; e < 4; ++e) {
        int j  = j0 + e;
        int dj = j - i;
        if (dj == 0) {
          vb[e] = xi; vl[e] = xi; vc[e] = xi;
        } else if (dj > 0) {
          bool m0 = dj <= 15;
          bool m1 = (dj >= 17) && (dj <= 31) && (dj & 1) && !(i & 1);
          bool m2 = (dj >= 35) && (((dj - 35) & 3) == 0) && !(i & 3);
          if (m0 || m1 || m2) {
            float xj = xr[j], xk = xr[(i + j) >> 1];
            vb[e] = (xi + xj) * 0.5f;
            vl[e] = (xi + xj + 0.5f * xk) * 0.4f;   // /2.5
            vc[e] = rm[idx + e];
          }
        }
      }
      vb *= wv; vl *= wv; vc *= wv;
      __builtin_nontemporal_store(vb, (v4f*)(outB + obase + idx));
      __builtin_nontemporal_store(vl, (v4f*)(outL + obase + idx));
      __builtin_nontemporal_store(vc, (v4f*)(outC + obase + idx));
    }
  }
}

// ---------------------------------------------------------------------------
// Kernel D: mask2d broadcast (B,1,64,64), 1.0/0.0 floats, b128 stores.
// ---------------------------------------------------------------------------
__global__ __launch_bounds__(256) void write_mask(float* __restrict__ mask)
{
  int g = blockIdx.x * 256 + threadIdx.x;   // 0..16383 groups of 4
  int e = (g & 1023) * 4;                   // element within one 64x64 slice
  int i = e >> 6, j0 = e & 63;
  v4f m4;
  #pragma unroll
  for (int e4 = 0; e4 < 4; ++e4) {
    int j  = j0 + e4;
    int dj = j - i;
    bool m = false;
    if (dj == 0) m = true;
    else if (dj > 0) {
      m = (dj <= 15) ||
          ((dj >= 17) && (dj <= 31) && (dj & 1) && !(i & 1)) ||
          ((dj >= 35) && (((dj - 35) & 3) == 0) && !(i & 3));
    }
    m4[e4] = m ? 1.0f : 0.0f;
  }
  __builtin_nontemporal_store(m4, (v4f*)(mask + (size_t)g * 4));
}

extern "C" void kernel_launch(void* const* d_in, const int* in_sizes, int n_in,
                              void* d_out, int out_size, void* d_ws, size_t ws_size,
                              hipStream_t stream) {
  (void)in_sizes; (void)n_in; (void)out_size; (void)ws_size;
  const float* x   = (const float*)d_in[0];   // (16,512,64)
  const float* w_c = (const float*)d_in[1];   // (1024,512)
  const float* b_c = (const float*)d_in[2];   // (1024,)
  const float* w_v = (const float*)d_in[3];   // (512,512)
  const float* b_v = (const float*)d_in[4];   // (512,)
  float* out = (float*)d_out;

  float* mkq = (float*)d_ws;                            // 16*64*1024 f32 = 4 MB
  float* wsm = mkq + (size_t)BATCH * NSEQ * OCAT;       // 16*64*64 f32

  gemm_kq<<<512, 256, 0, stream>>>(x, w_c, b_c, w_v, b_v, mkq);
  attn_softmax<<<BATCH, 128, 0, stream>>>(mkq, wsm);
  build_maps<<<BATCH * 64, 256, 0, stream>>>(x, wsm, out);

  const size_t mapstride = (size_t)BATCH * DDIM * NSEQ * NSEQ;
  write_mask<<<64, 256, 0, stream>>>(out + 3 * mapstride);
}